// EdgeInGTModel_7576322310512
// MI455X (gfx1250) — compile-verified
//
#include <hip/hip_runtime.h>
#include <math.h>

typedef __attribute__((ext_vector_type(2))) float v2f;
typedef __attribute__((ext_vector_type(8))) float v8f;

#define HIDD 128
#define NHD  8
#define NNOD 256
#define NBAT 16

// ---------------- generic single-wave fp32 WMMA GEMM (B column-major) -------
// C[M,N] = A[M,K] @ B[K,N], with B stored col-major: element(k,n) = Bt[n*ldb+k]
// grid.x = M/16, grid.y = N/(16*NT), grid.z = batch
// batch offsets: off = (z/div)*s1 + (z%div)*s2   for A, B, C (ADD shares C's)
// FUSE: 0 = none, 1 = relu, 2 = add ADD.  Requires K % 8 == 0.
template<int FUSE, int NT>
__global__ __launch_bounds__(32)
void wmma_gemm_f32(const float* __restrict__ A, const float* __restrict__ Bt,
                   float* __restrict__ C, const float* __restrict__ ADD,
                   int K, int lda, int ldb, int ldc, int div,
                   long long sA1, long long sA2,
                   long long sB1, long long sB2,
                   long long sC1, long long sC2)
{
    const int z  = blockIdx.z;
    const int zq = z / div, zr = z % div;
    const float* Ab = A  + zq * sA1 + zr * sA2;
    const float* Bb = Bt + zq * sB1 + zr * sB2;
    float*       Cb = C  + zq * sC1 + zr * sC2;
    const float* Db = (FUSE == 2) ? (ADD + zq * sC1 + zr * sC2) : nullptr;

    const int m0   = blockIdx.x * 16;
    const int n0   = blockIdx.y * (16 * NT);
    const int lane = threadIdx.x & 31;
    const int l16  = lane & 15;
    const int kh   = lane >> 4;          // K-half select (0/1)

    v8f acc[NT];
#pragma unroll
    for (int t = 0; t < NT; ++t)
#pragma unroll
        for (int r = 0; r < 8; ++r) acc[t][r] = 0.0f;

    // single lane-dependent base per matrix; tile index folded into constant
    // element offsets so all fragment loads become global_load_b64 base+imm.
    const float* __restrict__ paBase = Ab + (long long)(m0 + l16) * lda + 2 * kh;
    const float* __restrict__ pbBase = Bb + (long long)(n0 + l16) * ldb + 2 * kh;

    // 2-deep software pipeline: stage0 covers k0, stage1 covers k0+4
    v2f a0, a1, b0[NT], b1[NT];
    a0 = *(const v2f*)(paBase + 0);
#pragma unroll
    for (int t = 0; t < NT; ++t)
        b0[t] = *(const v2f*)(pbBase + (long long)t * 16 * ldb + 0);
    a1 = *(const v2f*)(paBase + 4);
#pragma unroll
    for (int t = 0; t < NT; ++t)
        b1[t] = *(const v2f*)(pbBase + (long long)t * 16 * ldb + 4);

    for (int k0 = 8; k0 < K; k0 += 8) {
#pragma unroll
        for (int t = 0; t < NT; ++t)
            acc[t] = __builtin_amdgcn_wmma_f32_16x16x4_f32(
                false, a0, false, b0[t], (short)0, acc[t], false, false);
        a0 = *(const v2f*)(paBase + k0);
#pragma unroll
        for (int t = 0; t < NT; ++t)
            b0[t] = *(const v2f*)(pbBase + (long long)t * 16 * ldb + k0);
#pragma unroll
        for (int t = 0; t < NT; ++t)
            acc[t] = __builtin_amdgcn_wmma_f32_16x16x4_f32(
                false, a1, false, b1[t], (short)0, acc[t], false, false);
        a1 = *(const v2f*)(paBase + k0 + 4);
#pragma unroll
        for (int t = 0; t < NT; ++t)
            b1[t] = *(const v2f*)(pbBase + (long long)t * 16 * ldb + k0 + 4);
    }
#pragma unroll
    for (int t = 0; t < NT; ++t)
        acc[t] = __builtin_amdgcn_wmma_f32_16x16x4_f32(
            false, a0, false, b0[t], (short)0, acc[t], false, false);
#pragma unroll
    for (int t = 0; t < NT; ++t)
        acc[t] = __builtin_amdgcn_wmma_f32_16x16x4_f32(
            false, a1, false, b1[t], (short)0, acc[t], false, false);

#pragma unroll
    for (int t = 0; t < NT; ++t) {
#pragma unroll
        for (int r = 0; r < 8; ++r) {
            const int row = m0 + r + 8 * kh;
            const int col = n0 + t * 16 + l16;
            float v = acc[t][r];
            if (FUSE == 1) v = fmaxf(v, 0.0f);
            if (FUSE == 2) v += Db[(long long)row * ldc + col];
            Cb[(long long)row * ldc + col] = v;
        }
    }
}

// ---------------- layernorm: one wave per 128-wide row ----------------
__global__ __launch_bounds__(256)
void ln_kernel(const float* __restrict__ x, const float* __restrict__ res,
               const float* __restrict__ g, const float* __restrict__ b,
               float* __restrict__ out, int rows)
{
    const int wave = threadIdx.x >> 5;
    const int lane = threadIdx.x & 31;
    const int row  = blockIdx.x * 8 + wave;
    if (row >= rows) return;
    const long long base = (long long)row * HIDD;
    float v[4];
    float s = 0.0f, s2 = 0.0f;
#pragma unroll
    for (int i = 0; i < 4; ++i) {
        const int c = lane + 32 * i;
        float t = x[base + c];
        if (res) t += res[base + c];
        v[i] = t; s += t; s2 += t * t;
    }
#pragma unroll
    for (int o = 16; o >= 1; o >>= 1) {
        s  += __shfl_xor(s,  o, 32);
        s2 += __shfl_xor(s2, o, 32);
    }
    const float mu   = s * (1.0f / HIDD);
    const float var  = s2 * (1.0f / HIDD) - mu * mu;
    const float rinv = rsqrtf(var + 1e-5f);
#pragma unroll
    for (int i = 0; i < 4; ++i) {
        const int c = lane + 32 * i;
        out[base + c] = (v[i] - mu) * rinv * g[c] + b[c];
    }
}

// ------ fused scale + edge-bias + softmax + edge-gate (in place on att) ------
__global__ __launch_bounds__(256)
void softmax_kernel(float* __restrict__ att, const float* __restrict__ ef,
                    const float* __restrict__ comb)
{
    __shared__ float red[8];
    const int i  = blockIdx.x;
    const int bh = blockIdx.y;
    const int b  = bh >> 3;
    const int h  = bh & 7;
    const int j  = threadIdx.x;
    const int wave = threadIdx.x >> 5;
    const int lane = threadIdx.x & 31;

    const float cB = comb[h];
    const float cG = comb[NHD + h];
    const float efv = ef[((long long)(b * NNOD + i)) * NNOD + j];
    const long long aidx = (long long)bh * NNOD * NNOD + (long long)i * NNOD + j;

    float x = att[aidx] * 0.25f + efv * cB;   // 1/sqrt(HD)=0.25

    float m = x;
#pragma unroll
    for (int o = 16; o >= 1; o >>= 1) m = fmaxf(m, __shfl_xor(m, o, 32));
    if (lane == 0) red[wave] = m;
    __syncthreads();
    float mm = red[lane & 7];
#pragma unroll
    for (int o = 4; o >= 1; o >>= 1) mm = fmaxf(mm, __shfl_xor(mm, o, 32));
    __syncthreads();

    const float p = __expf(x - mm);

    float s = p;
#pragma unroll
    for (int o = 16; o >= 1; o >>= 1) s += __shfl_xor(s, o, 32);
    if (lane == 0) red[wave] = s;
    __syncthreads();
    float ss = red[lane & 7];
#pragma unroll
    for (int o = 4; o >= 1; o >>= 1) ss += __shfl_xor(ss, o, 32);

    att[aidx] = (p / ss) * efv * cG;
}

// ---------------- tiny kernels ----------------
__global__ void comb_kernel(const float* __restrict__ We_in,
                            const float* __restrict__ We,
                            float* __restrict__ comb)
{
    const int t = threadIdx.x;
    if (t >= 48) return;
    const int l = t >> 4, c = t & 15;
    float s = 0.0f;
    for (int d = 0; d < HIDD; ++d)
        s += We_in[d] * We[((long long)l * HIDD + d) * 16 + c];
    comb[t] = s;
}

__global__ void h0_kernel(const float* __restrict__ nf, const float* __restrict__ Wn,
                          float* __restrict__ h)
{
    const long long idx = (long long)blockIdx.x * blockDim.x + threadIdx.x;
    if (idx >= (long long)NBAT * NNOD * HIDD) return;
    const long long row = idx / HIDD;
    const int c = (int)(idx % HIDD);
    h[idx] = nf[row * 2] * Wn[c] + nf[row * 2 + 1] * Wn[HIDD + c];
}

// out[l][n][m] = in[l][m][n]   (L batched M x N -> N x M transpose)
__global__ void transpose_kernel(const float* __restrict__ in, float* __restrict__ out,
                                 int L, int M, int N)
{
    const long long idx = (long long)blockIdx.x * blockDim.x + threadIdx.x;
    const long long tot = (long long)L * M * N;
    if (idx >= tot) return;
    const long long mn = (long long)M * N;
    const int l = (int)(idx / mn);
    const int r = (int)(idx % mn);
    const int m = r / N;
    const int n = r % N;
    out[(long long)l * mn + (long long)n * M + m] = in[idx];
}

// vt[b][c][j] = qkv[b][j][256 + c]   (V transposed per batch, c in [0,128))
__global__ void vtrans_kernel(const float* __restrict__ qkv, float* __restrict__ vt)
{
    const long long idx = (long long)blockIdx.x * blockDim.x + threadIdx.x;
    if (idx >= (long long)NBAT * HIDD * NNOD) return;
    const int b = (int)(idx / (HIDD * NNOD));
    const int r = (int)(idx % (HIDD * NNOD));
    const int c = r / NNOD;
    const int j = r % NNOD;
    vt[idx] = qkv[((long long)(b * NNOD + j)) * 384 + 256 + c];
}

__global__ void dec_kernel(const float* __restrict__ h, const float* __restrict__ Wdec,
                           float* __restrict__ out)
{
    const int row = blockIdx.x * blockDim.x + threadIdx.x;
    if (row >= NBAT * NNOD) return;
    float s = 0.0f;
    for (int d = 0; d < HIDD; ++d) s += h[(long long)row * HIDD + d] * Wdec[d];
    out[row] = 10.0f * tanhf(s * 0.08838834764831845f);  // 1/sqrt(128)
}

// ---------------- host launch ----------------
extern "C" void kernel_launch(void* const* d_in, const int* in_sizes, int n_in,
                              void* d_out, int out_size, void* d_ws, size_t ws_size,
                              hipStream_t stream)
{
    const float* nf    = (const float*)d_in[0];   // (16,256,2)
    const float* ef    = (const float*)d_in[1];   // (16,256,256,1)
    const float* Wn    = (const float*)d_in[2];   // (2,128)
    const float* We_in = (const float*)d_in[3];   // (1,128)
    const float* ln1g  = (const float*)d_in[4];   // (3,128)
    const float* ln1b  = (const float*)d_in[5];
    const float* Wh    = (const float*)d_in[6];   // (3,128,384)
    const float* We    = (const float*)d_in[7];   // (3,128,16)
    const float* ln2g  = (const float*)d_in[8];
    const float* ln2b  = (const float*)d_in[9];
    const float* W1    = (const float*)d_in[10];  // (3,128,512)
    const float* W2    = (const float*)d_in[11];  // (3,512,128)
    const float* Wdec  = (const float*)d_in[12];  // (128,1)
    float* out = (float*)d_out;

    float* ws = (float*)d_ws;
    const long long N = NNOD;
    float* h    = ws;                         // 524288
    float* hn   = ws + 524288;                // 524288
    float* qkv  = ws + 1048576;               // 1572864
    float* att  = ws + 2621440;               // 8388608
    float* y    = ws + 11010048;              // 524288
    float* z    = ws + 11534336;              // 524288
    float* act  = ws + 12058624;              // 2097152
    float* comb = ws + 14155776;              // 64 (48 used)
    float* whT  = ws + 14155840;              // 3*384*128 = 147456
    float* w1T  = ws + 14303296;              // 3*512*128 = 196608
    float* w2T  = ws + 14499904;              // 3*128*512 = 196608
    float* vt   = ws + 14696512;              // 524288

    comb_kernel<<<1, 64, 0, stream>>>(We_in, We, comb);
    h0_kernel<<<(NBAT * NNOD * HIDD + 255) / 256, 256, 0, stream>>>(nf, Wn, h);

    // pre-transpose weights to column-major for contiguous b64 fragment loads
    transpose_kernel<<<(3 * 128 * 384 + 255) / 256, 256, 0, stream>>>(Wh, whT, 3, 128, 384);
    transpose_kernel<<<(3 * 128 * 512 + 255) / 256, 256, 0, stream>>>(W1, w1T, 3, 128, 512);
    transpose_kernel<<<(3 * 512 * 128 + 255) / 256, 256, 0, stream>>>(W2, w2T, 3, 512, 128);

    for (int l = 0; l < 3; ++l) {
        // hn = LN1(h)
        ln_kernel<<<NBAT * NNOD / 8, 256, 0, stream>>>(
            h, nullptr, ln1g + l * HIDD, ln1b + l * HIDD, hn, NBAT * NNOD);

        // qkv = hn @ Wh[l]   (M=256,K=128,N=384 per batch), B = whT[l] (384x128)
        wmma_gemm_f32<0, 4><<<dim3(16, 6, 16), 32, 0, stream>>>(
            hn, whT + (long long)l * 384 * HIDD, qkv, nullptr,
            128, 128, 128, 384, 1, N * 128, 0, 0, 0, N * 384, 0);

        // scores = q @ k^T per (b,h): B(col=j,k=d) = qkv[j*384 + 128 + h*16 + d]
        wmma_gemm_f32<0, 4><<<dim3(16, 4, 128), 32, 0, stream>>>(
            qkv, qkv + 128, att, nullptr,
            16, 384, 384, 256, 8,
            N * 384, 16, N * 384, 16, (long long)8 * 65536, 65536);

        // att = softmax(att*0.25 + ef*combB) * ef * combG
        softmax_kernel<<<dim3(NNOD, NBAT * NHD), 256, 0, stream>>>(
            att, ef, comb + l * 16);

        // repack V transposed: vt[b][c][j]
        vtrans_kernel<<<(NBAT * HIDD * NNOD + 255) / 256, 256, 0, stream>>>(qkv, vt);

        // y[b,i,h*16+d] = att @ v per (b,h): B(col=d,k=j) = vt[b][h*16+d][j]
        wmma_gemm_f32<0, 1><<<dim3(16, 1, 128), 32, 0, stream>>>(
            att, vt, y, nullptr,
            256, 256, 256, 128, 8,
            (long long)8 * 65536, 65536, (long long)HIDD * NNOD, 16 * NNOD,
            N * 128, 16);

        // z = LN2(y + h)
        ln_kernel<<<NBAT * NNOD / 8, 256, 0, stream>>>(
            y, h, ln2g + l * HIDD, ln2b + l * HIDD, z, NBAT * NNOD);

        // act = relu(z @ W1[l])   (M=256,K=128,N=512), B = w1T[l] (512x128)
        wmma_gemm_f32<1, 4><<<dim3(16, 8, 16), 32, 0, stream>>>(
            z, w1T + (long long)l * 512 * HIDD, act, nullptr,
            128, 128, 128, 512, 1, N * 128, 0, 0, 0, N * 512, 0);

        // h = act @ W2[l] + y   (M=256,K=512,N=128), B = w2T[l] (128x512)
        wmma_gemm_f32<2, 4><<<dim3(16, 2, 16), 32, 0, stream>>>(
            act, w2T + (long long)l * HIDD * 512, h, y,
            512, 512, 512, 128, 1, N * 512, 0, 0, 0, N * 128, 0);
    }

    // out = 10 * tanh((h @ Wdec) / sqrt(128))
    dec_kernel<<<(NBAT * NNOD + 255) / 256, 256, 0, stream>>>(h, Wdec, out);
}